// ReductionAndExpansionAreaResamp_57861799411998
// MI455X (gfx1250) — compile-verified
//
#include <hip/hip_runtime.h>
#include <hip/hip_bf16.h>
#include <math.h>

// ---------------- problem constants (static shapes from reference) ----------
#define B_     8
#define L_IN_  4096
#define D_     256
#define MAX_T_ 2048
#define TTILES (MAX_T_/16)   // 128 t-tiles of 16 bins
#define WINMAX 68            // max input rows overlapped by a 16-bin tile (<=66, padded to mult of 4)
#define DCOLS  128           // columns of D staged per block (half of D)
#define NOISE_SCALE 0.1f
#define EPS_ 1e-12f

typedef __attribute__((ext_vector_type(2))) float v2f;
typedef __attribute__((ext_vector_type(8))) float v8f;
typedef __attribute__((ext_vector_type(4))) int   v4i;

#define AS1 __attribute__((address_space(1)))
#define AS3 __attribute__((address_space(3)))

#if defined(__has_builtin)
#if __has_builtin(__builtin_amdgcn_global_load_async_to_lds_b128)
#define HAVE_ASYNC_LDS 1
#endif
#endif

// ---------------- threefry2x32 -> normal, matching jax.random.normal(key(42))
__device__ __forceinline__ unsigned int rotl32(unsigned int x, int n) {
  return (x << n) | (x >> (32 - n));
}

__device__ __forceinline__ float tf_normal(unsigned int idx) {
  // jax random_bits: counts = iota(N); x0 = counts[:N/2], x1 = counts[N/2:]
  const unsigned int HALFN = (unsigned int)(B_ * MAX_T_ * D_) / 2u; // 2097152
  unsigned int x0, x1; int pick;
  if (idx < HALFN) { x0 = idx;         x1 = idx + HALFN; pick = 0; }
  else             { x0 = idx - HALFN; x1 = idx;         pick = 1; }
  const unsigned int k0 = 0u, k1 = 42u;                 // jax.random.key(42)
  const unsigned int k2 = 0x1BD11BDAu ^ k0 ^ k1;
  const int R0[4] = {13, 15, 26, 6};
  const int R1[4] = {17, 29, 16, 24};
  x0 += k0; x1 += k1;
#pragma unroll
  for (int g = 0; g < 5; g++) {
    const int* R = (g & 1) ? R1 : R0;
#pragma unroll
    for (int r = 0; r < 4; r++) { x0 += x1; x1 = rotl32(x1, R[r]); x1 ^= x0; }
    switch (g) {
      case 0: x0 += k1; x1 += k2 + 1u; break;
      case 1: x0 += k2; x1 += k0 + 2u; break;
      case 2: x0 += k0; x1 += k1 + 3u; break;
      case 3: x0 += k1; x1 += k2 + 4u; break;
      case 4: x0 += k2; x1 += k0 + 5u; break;
    }
  }
  unsigned int bits = pick ? x1 : x0;
  unsigned int fb = (bits >> 9) | 0x3f800000u;          // [1,2)
  float u = __uint_as_float(fb) - 1.0f;                 // [0,1)
  const float lo = -0.99999994f;                        // nextafter(-1,0)
  float val = u * (1.0f - lo) + lo;
  val = fmaxf(val, lo);
  return 1.41421356f * erfinvf(val);                    // sqrt(2)*erfinv(u)
}

// ---------------- kernel A: stable partition (== stable argsort of pad mask) -
__global__ __launch_bounds__(256) void compact_kernel(
    const unsigned char* __restrict__ pmask, int* __restrict__ order,
    int* __restrict__ Lbuf) {
  __shared__ int cnts[256];
  __shared__ int excl[257];
  const int b = blockIdx.x, t = threadIdx.x;
  const unsigned char* mrow = pmask + (long)b * L_IN_;
  const int base = t * 16;
  int c = 0;
#pragma unroll
  for (int k = 0; k < 16; k++) c += (mrow[base + k] == 0);
  cnts[t] = c;
  __syncthreads();
  if (t == 0) {
    int s = 0;
    for (int i = 0; i < 256; i++) { excl[i] = s; s += cnts[i]; }
    excl[256] = s;
  }
  __syncthreads();
  const int L = excl[256];
  if (t == 0) Lbuf[b] = L;
  int vrun = excl[t];
  int* orow = order + (long)b * L_IN_;
  for (int k = 0; k < 16; k++) {
    const int p = base + k;
    if (mrow[p] == 0) { orow[vrun] = p; vrun++; }
    else              { orow[L + (p - vrun)] = p; }   // pads keep order at back
  }
}

// ---------------- kernel B: banded WMMA resample + noise ---------------------
__global__ __launch_bounds__(256) void resample_kernel(
    const float* __restrict__ x, const int* __restrict__ finallength,
    const int* __restrict__ order, const int* __restrict__ Lbuf,
    float* __restrict__ out, unsigned char* __restrict__ outMask) {
  __shared__ float smem[WINMAX * DCOLS];   // 34 KB staged input window

  const int bid   = blockIdx.x;
  const int dHalf = bid & 1;
  const int tTile = (bid >> 1) & (TTILES - 1);
  const int b     = bid >> 8;              // 256 blocks per batch
  const int t0    = tTile * 16;
  const int d0b   = dHalf * DCOLS;

  const int L  = Lbuf[b];
  const int T  = finallength[b];
  const int Tc = (T < 1) ? 1 : T;

  // input-row window covered by bins [t0, t0+16): [iLo, iHi)
  int iLo = (int)(((long)t0 * L) / Tc);
  int iHi = (int)(((long)(t0 + 16) * L + Tc - 1) / Tc);
  if (iHi > L) iHi = L;
  int win = iHi - iLo;
  if (win < 0) win = 0;
  if (win > WINMAX) win = WINMAX;          // safety (cannot trigger: <=66)
  const int winPad = (win + 3) & ~3;       // pad rows are zero-filled below

  const int tid = threadIdx.x;

  // ---- stage window rows (gather via stable-partition order) --------------
#ifdef HAVE_ASYNC_LDS
  // CDNA5 async global->LDS: no VGPR round trip, tracked by ASYNCcnt
  for (int idx = tid; idx < win * (DCOLS / 4); idx += 256) {
    const int r = idx >> 5, c = idx & 31;
    const int src = order[(long)b * L_IN_ + iLo + r];
    AS1 v4i* gp = (AS1 v4i*)(x + (((long)b * L_IN_ + src) * D_ + d0b + c * 4));
    AS3 v4i* lp = (AS3 v4i*)(&smem[r * DCOLS + c * 4]);
    __builtin_amdgcn_global_load_async_to_lds_b128(gp, lp, 0, 0);
  }
  // zero-fill pad rows [win, winPad) so the WMMA k-loop needs no bounds check
  for (int idx = win * (DCOLS / 4) + tid; idx < winPad * (DCOLS / 4); idx += 256) {
    const float4 z = {0.f, 0.f, 0.f, 0.f};
    *(float4*)(&smem[idx * 4]) = z;
  }
#if __has_builtin(__builtin_amdgcn_s_wait_asynccnt)
  __builtin_amdgcn_s_wait_asynccnt(0);
#else
  asm volatile("s_wait_asynccnt 0x0" ::: "memory");
#endif
#else
  for (int idx = tid; idx < winPad * (DCOLS / 4); idx += 256) {
    const int r = idx >> 5, c = idx & 31;
    float4 v = {0.f, 0.f, 0.f, 0.f};
    if (r < win) {
      const int src = order[(long)b * L_IN_ + iLo + r];
      v = *(const float4*)(x + (((long)b * L_IN_ + src) * D_ + d0b + c * 4));
    }
    *(float4*)(&smem[r * DCOLS + c * 4]) = v;
  }
#endif
  __syncthreads();

  const int wave = tid >> 5, lane = tid & 31;
  const int half = lane >> 4, m = lane & 15;
  const int cbase = wave * 16 + m;         // this lane's column inside smem row
  const int dcol  = d0b + cbase;           // global feature column

  // per-lane A-row parameters (row = t0+m is fixed per lane)
  const int tRow = t0 + m;
  const bool vRow = (tRow < T) && (L > 0);
  const long jl = (long)tRow * L;
  const int i0r = (int)(jl / Tc);
  const int i1r = (int)((jl + L + Tc - 1) / Tc);
  int cnt = i1r - i0r; if (cnt < 1) cnt = 1;
  const float wint = vRow ? (1.0f / (float)cnt) : 0.0f;
  const float Lf = (float)L;
  const float step = Lf / (float)Tc;
  const float startf = (float)tRow * step;
  const float endf = startf + step;

  v8f accM = {0.f,0.f,0.f,0.f,0.f,0.f,0.f,0.f};   // area pool
  v8f accU = {0.f,0.f,0.f,0.f,0.f,0.f,0.f,0.f};   // frac-weighted sum(x)
  v8f accQ = {0.f,0.f,0.f,0.f,0.f,0.f,0.f,0.f};   // frac-weighted sum(x^2)

  // running input index for this lane's K pair (A layout: lane half -> K 0,1 / 2,3)
  int   iBase  = iLo + half * 2;
  float fiBase = (float)iBase;             // hoist int->float out of the loop
  const float* srow = &smem[half * 2 * DCOLS + cbase];

  const int ksteps = winPad >> 2;
  for (int ks = 0; ks < ksteps; ks++) {
    v2f aI, aF, bV, bQ;
#pragma unroll
    for (int j = 0; j < 2; j++) {
      const int i = iBase + j;
      aI[j] = (i >= i0r && i < i1r) ? wint : 0.0f;
      const float fi = fiBase + (float)j;
      float w = fminf(fi + 1.0f, endf) - fmaxf(fi, startf);
      w = fmaxf(w, 0.0f);
      if (!vRow || i >= L) w = 0.0f;
      aF[j] = w;
      const float v = srow[(ks * 4 + j) * DCOLS];   // unconditional: window zero-padded
      bV[j] = v;
      bQ[j] = v * v;
    }
    // D = A x B + C  (full-precision f32 WMMA, 16x16x4)
    accM = __builtin_amdgcn_wmma_f32_16x16x4_f32(false, aI, false, bV, (short)0, accM, false, false);
    accU = __builtin_amdgcn_wmma_f32_16x16x4_f32(false, aF, false, bV, (short)0, accU, false, false);
    accQ = __builtin_amdgcn_wmma_f32_16x16x4_f32(false, aF, false, bQ, (short)0, accQ, false, false);
    iBase  += 4;
    fiBase += 4.0f;
  }

  // C/D layout: VGPR r holds row t0 + r (+8 for lanes 16-31), col = d0b+wave*16+(lane&15)
  float* outB = out + (long)b * MAX_T_ * D_;
#pragma unroll
  for (int r = 0; r < 8; r++) {
    const int t = t0 + half * 8 + r;
    const bool valid = (t < T) && (L > 0);
    float o = 0.0f;
    if (valid) {
      const float s0 = (float)t * step;
      const float e0 = s0 + step;
      float ws = fminf(e0, Lf) - s0;
      ws = fmaxf(ws, EPS_);
      const float inv = 1.0f / ws;
      const float mean = accU[r] * inv;
      const float msq  = accQ[r] * inv;
      const float var  = fmaxf(msq - mean * mean, EPS_);
      const float sd   = sqrtf(var);
      const unsigned int nidx = ((unsigned int)(b * MAX_T_ + t)) * D_ + (unsigned int)dcol;
      o = accM[r] + tf_normal(nidx) * sd * NOISE_SCALE;
    }
    outB[(long)t * D_ + dcol] = o;
    if (dHalf == 0 && wave == 0 && m == 0)
      outMask[b * MAX_T_ + t] = valid ? 0 : 1;   // True = pad
  }
}

// ---------------- host shim -------------------------------------------------
extern "C" void kernel_launch(void* const* d_in, const int* in_sizes, int n_in,
                              void* d_out, int out_size, void* d_ws, size_t ws_size,
                              hipStream_t stream) {
  const float*         x     = (const float*)d_in[0];
  const int*           flen  = (const int*)d_in[1];
  const unsigned char* pmask = (const unsigned char*)d_in[2];
  (void)in_sizes; (void)n_in; (void)out_size; (void)ws_size;

  int* Lbuf  = (int*)d_ws;                    // 8 ints
  int* order = (int*)d_ws + 8;                // 8*4096 ints

  float* out = (float*)d_out;
  unsigned char* outMask =
      (unsigned char*)d_out + (size_t)B_ * MAX_T_ * D_ * sizeof(float);

  compact_kernel<<<B_, 256, 0, stream>>>(pmask, order, Lbuf);
  resample_kernel<<<B_ * TTILES * 2, 256, 0, stream>>>(x, flen, order, Lbuf, out, outMask);
}